// Motion_Grid_75771813036385
// MI455X (gfx1250) — compile-verified
//
#include <hip/hip_runtime.h>

typedef __attribute__((ext_vector_type(2))) float v2f;
typedef __attribute__((ext_vector_type(8))) float v8f;

#define NPTS 2097152

// Workspace layout (float offsets), all 16B aligned.
#define GV0_OFF 0         // 4ch * 32^3  = 131072 floats
#define GV1_OFF 131072    // 4ch * 64^3  = 1048576 floats
#define GV2_OFF 1179648   // 2ch * 128^3 = 4194304 floats

// ---- prep: interleave channels of (grid + noise) into ws ----
__global__ void prep_kernel(const float* __restrict__ g, const float* __restrict__ nz,
                            float* __restrict__ dst, int vox, int C) {
    int t = blockIdx.x * blockDim.x + threadIdx.x;
    if (t >= vox * C) return;
    int v = t / C;
    int c = t - v * C;
    dst[t] = g[c * vox + v] + nz[c * vox + v];
}

// Trilinear interp on channel-interleaved volume, border weights zeroed (matches reference).
template <int C>
__device__ __forceinline__ void trilerp(const float* __restrict__ vol, int R,
                                        float cx, float cy, float cz,
                                        float* __restrict__ out) {
    float xs = (cx + 1.0f) * (0.5f * (float)R) - 0.5f;
    float ys = (cy + 1.0f) * (0.5f * (float)R) - 0.5f;
    float zs = (cz + 1.0f) * (0.5f * (float)R) - 0.5f;
    float x0f = floorf(xs), y0f = floorf(ys), z0f = floorf(zs);
    float tx = xs - x0f, ty = ys - y0f, tz = zs - z0f;
    int x0 = (int)x0f, y0 = (int)y0f, z0 = (int)z0f;
    float wx[2] = {1.0f - tx, tx};
    float wy[2] = {1.0f - ty, ty};
    float wz[2] = {1.0f - tz, tz};
#pragma unroll
    for (int c = 0; c < C; ++c) out[c] = 0.0f;
#pragma unroll
    for (int dz = 0; dz < 2; ++dz) {
#pragma unroll
        for (int dy = 0; dy < 2; ++dy) {
#pragma unroll
            for (int dx = 0; dx < 2; ++dx) {
                int xi = x0 + dx, yi = y0 + dy, zi = z0 + dz;
                bool valid = ((unsigned)xi < (unsigned)R) & ((unsigned)yi < (unsigned)R) &
                             ((unsigned)zi < (unsigned)R);
                float w = wz[dz] * wy[dy] * wx[dx];
                w = valid ? w : 0.0f;
                int xc = min(max(xi, 0), R - 1);
                int yc = min(max(yi, 0), R - 1);
                int zc = min(max(zi, 0), R - 1);
                int idx = ((zc * R + yc) * R + xc) * C;
                if constexpr (C == 4) {
                    const float4 v = *reinterpret_cast<const float4*>(vol + idx);
                    out[0] = fmaf(w, v.x, out[0]);
                    out[1] = fmaf(w, v.y, out[1]);
                    out[2] = fmaf(w, v.z, out[2]);
                    out[3] = fmaf(w, v.w, out[3]);
                } else {
                    const float2 v = *reinterpret_cast<const float2*>(vol + idx);
                    out[0] = fmaf(w, v.x, out[0]);
                    out[1] = fmaf(w, v.y, out[1]);
                }
            }
        }
    }
}

__global__ void __launch_bounds__(256)
motion_grid_kernel(const float* __restrict__ x,
                   const float* __restrict__ w1, const float* __restrict__ b1,
                   const float* __restrict__ w2, const float* __restrict__ b2,
                   const float* __restrict__ ws, float* __restrict__ out) {
    __shared__ float f_lds[8][32][20];   // features, per wave: 32 points x 20
    __shared__ float h_lds[8][16][65];   // hidden,   per wave: 16 points x 64 (pad 65)

    const int tid  = threadIdx.x;
    const int wave = tid >> 5;
    const int lane = tid & 31;
    const int lh   = lane >> 4;   // K-pair half select
    const int ln   = lane & 15;   // N / M-row select

    const int p = blockIdx.x * 256 + tid;

    // ---------------- feature phase: 1 point / thread ----------------
    const float xv0 = x[p * 3 + 0];
    const float xv1 = x[p * 3 + 1];
    const float xv2 = x[p * 3 + 2];

    float feat[20];
    const float PI_F = 3.14159265358979323846f;
    {
        float s0, c0, s1, c1, s2, c2;
        __sincosf(PI_F * xv0, &s0, &c0);
        __sincosf(PI_F * xv1, &s1, &c1);
        __sincosf(PI_F * xv2, &s2, &c2);
        trilerp<4>(ws + GV0_OFF, 32, s0, s1, s2, &feat[0]);
        trilerp<4>(ws + GV0_OFF, 32, c0, c1, c2, &feat[4]);
    }
    {
        float s0, c0, s1, c1, s2, c2;
        __sincosf(2.0f * PI_F * xv0, &s0, &c0);
        __sincosf(2.0f * PI_F * xv1, &s1, &c1);
        __sincosf(2.0f * PI_F * xv2, &s2, &c2);
        trilerp<4>(ws + GV1_OFF, 64, s0, s1, s2, &feat[8]);
        trilerp<4>(ws + GV1_OFF, 64, c0, c1, c2, &feat[12]);
    }
    {
        float s0, c0, s1, c1, s2, c2;
        __sincosf(4.0f * PI_F * xv0, &s0, &c0);
        __sincosf(4.0f * PI_F * xv1, &s1, &c1);
        __sincosf(4.0f * PI_F * xv2, &s2, &c2);
        trilerp<2>(ws + GV2_OFF, 128, s0, s1, s2, &feat[16]);
        trilerp<2>(ws + GV2_OFF, 128, c0, c1, c2, &feat[18]);
    }
#pragma unroll
    for (int k = 0; k < 20; ++k) f_lds[wave][lane][k] = feat[k];
    __syncthreads();

    // ---------------- preload B fragments (weights) ----------------
    // B-matrix 4x16 f32 layout: VGPR j, lane: K = kt*4 + j + 2*lh, N = ln.
    v2f b1f[5][4];
#pragma unroll
    for (int kt = 0; kt < 5; ++kt) {
#pragma unroll
        for (int nt = 0; nt < 4; ++nt) {
            const int K0 = kt * 4 + 2 * lh;
            const int N  = nt * 16 + ln;
            b1f[kt][nt].x = w1[N * 20 + K0];
            b1f[kt][nt].y = w1[N * 20 + K0 + 1];
        }
    }
    v2f b2f[16];
#pragma unroll
    for (int kt = 0; kt < 16; ++kt) {
        const int K0 = kt * 4 + 2 * lh;
        b2f[kt].x = (ln < 7) ? w2[ln * 64 + K0] : 0.0f;
        b2f[kt].y = (ln < 7) ? w2[ln * 64 + K0 + 1] : 0.0f;
    }
    float b1v[4];
#pragma unroll
    for (int nt = 0; nt < 4; ++nt) b1v[nt] = b1[nt * 16 + ln];
    const float b2v = (ln < 7) ? b2[ln] : 0.0f;

    const int pB = blockIdx.x * 256 + wave * 32;

    // ---------------- two 16-point M-tiles per wave ----------------
#pragma unroll
    for (int t = 0; t < 2; ++t) {
        // A fragments layer 1 (16x4 f32: lane=M row, lane-half picks K pair)
        v2f a1[5];
#pragma unroll
        for (int kt = 0; kt < 5; ++kt) {
            const int K0 = kt * 4 + 2 * lh;
            a1[kt].x = f_lds[wave][t * 16 + ln][K0];
            a1[kt].y = f_lds[wave][t * 16 + ln][K0 + 1];
        }
        // Layer 1: 16x20 @ 20x64, K in steps of 4, N in 4 tiles of 16.
        v8f acc[4];
#pragma unroll
        for (int nt = 0; nt < 4; ++nt) {
            v8f c = {};
#pragma unroll
            for (int kt = 0; kt < 5; ++kt) {
                c = __builtin_amdgcn_wmma_f32_16x16x4_f32(
                        false, a1[kt], false, b1f[kt][nt], (short)0, c, false, false);
            }
            acc[nt] = c;
        }
        // bias + leaky ReLU, stage h via LDS (C-layout -> A-layout transpose)
#pragma unroll
        for (int nt = 0; nt < 4; ++nt) {
#pragma unroll
            for (int r = 0; r < 8; ++r) {
                float v = acc[nt][r] + b1v[nt];
                v = v > 0.0f ? v : 0.01f * v;
                h_lds[wave][r + 8 * lh][nt * 16 + ln] = v;  // row M, col N
            }
        }
        // Layer 2: 16x64 @ 64x16 (cols 7..15 are zero-padded weights)
        v8f acc2 = {};
#pragma unroll
        for (int kt = 0; kt < 16; ++kt) {
            const int K0 = kt * 4 + 2 * lh;
            v2f a2;
            a2.x = h_lds[wave][ln][K0];
            a2.y = h_lds[wave][ln][K0 + 1];
            acc2 = __builtin_amdgcn_wmma_f32_16x16x4_f32(
                       false, a2, false, b2f[kt], (short)0, acc2, false, false);
        }
        // store D (C/D layout: VGPR r -> row r + 8*lh, col ln), cols 0..6 valid
        if (ln < 7) {
#pragma unroll
            for (int r = 0; r < 8; ++r) {
                const int row = pB + t * 16 + r + 8 * lh;
                out[row * 7 + ln] = acc2[r] + b2v;
            }
        }
    }
}

extern "C" void kernel_launch(void* const* d_in, const int* in_sizes, int n_in,
                              void* d_out, int out_size, void* d_ws, size_t ws_size,
                              hipStream_t stream) {
    (void)in_sizes; (void)n_in; (void)out_size; (void)ws_size;
    const float* x  = (const float*)d_in[0];
    const float* g0 = (const float*)d_in[1];
    const float* g1 = (const float*)d_in[2];
    const float* g2 = (const float*)d_in[3];
    const float* n0 = (const float*)d_in[4];
    const float* n1 = (const float*)d_in[5];
    const float* n2 = (const float*)d_in[6];
    const float* w1 = (const float*)d_in[7];
    const float* b1 = (const float*)d_in[8];
    const float* w2 = (const float*)d_in[9];
    const float* b2 = (const float*)d_in[10];
    float* ws  = (float*)d_ws;
    float* out = (float*)d_out;

    // interleave (grid + noise) once per launch into L2-resident tables
    prep_kernel<<<(131072  + 255) / 256, 256, 0, stream>>>(g0, n0, ws + GV0_OFF, 32768,   4);
    prep_kernel<<<(1048576 + 255) / 256, 256, 0, stream>>>(g1, n1, ws + GV1_OFF, 262144,  4);
    prep_kernel<<<(4194304 + 255) / 256, 256, 0, stream>>>(g2, n2, ws + GV2_OFF, 2097152, 2);

    // 2,097,152 points / 256 per block
    motion_grid_kernel<<<NPTS / 256, 256, 0, stream>>>(x, w1, b1, w2, b2, ws, out);
}